// Node_Generator_29008209117364
// MI455X (gfx1250) — compile-verified
//
#include <hip/hip_runtime.h>
#include <hip/hip_bf16.h>

// Problem constants from the reference
#define N_NODES 100000
#define N_EDGES 1600000
#define F_DIM   512
#define H_DIM   256

typedef float v2f __attribute__((ext_vector_type(2)));
typedef float v8f __attribute__((ext_vector_type(8)));

// ---------------------------------------------------------------- utilities
__global__ void zero_kernel(float* __restrict__ p, size_t n) {
    size_t i = (size_t)blockIdx.x * blockDim.x + threadIdx.x;
    if (i < n) p[i] = 0.0f;
}

// degrees via float atomics (avg 16 edges/node -> low contention)
__global__ void deg_kernel(const int* __restrict__ src, const int* __restrict__ dst,
                           float* __restrict__ deg_src, float* __restrict__ deg_dst) {
    int i = blockIdx.x * blockDim.x + threadIdx.x;
    if (i < N_EDGES) {
        atomicAdd(&deg_src[src[i]], 1.0f);
        atomicAdd(&deg_dst[dst[i]], 1.0f);
    }
}

__global__ void norm_kernel(float* __restrict__ a, float* __restrict__ b) {
    int i = blockIdx.x * blockDim.x + threadIdx.x;
    if (i < N_NODES) {
        a[i] = rsqrtf(fmaxf(a[i], 1.0f));
        b[i] = rsqrtf(fmaxf(b[i], 1.0f));
    }
}

// ------------------------------------------------- fp32 WMMA tiled GEMM
// Y[M x 256] = (X * scale[:,None]) @ W[K x 256]
// One wave per 16x16 output tile, V_WMMA_F32_16X16X4_F32 over the K loop.
// A layout: lane<16 -> M=lane, K pair {0,1}; lane>=16 -> same M, K pair {2,3}
// B layout mirrors with N across lanes; C/D: acc[r] <-> M = r + 8*(lane>>4).
__global__ __launch_bounds__(128)
void gemm_scaled_wmma(const float* __restrict__ X, const float* __restrict__ W,
                      const float* __restrict__ scale, float* __restrict__ Y,
                      int K) {
    const int wave = threadIdx.x >> 5;
    const int lane = threadIdx.x & 31;
    const int gtile = blockIdx.x * 4 + wave;          // 0 .. (M/16)*(H/16)-1
    const int tile_m = (gtile >> 4) << 4;             // H/16 == 16 column tiles
    const int tile_n = (gtile & 15) << 4;
    const int hf  = lane >> 4;                        // 0 or 1 (K sub-pair)
    const int idx = lane & 15;                        // M for A, N for B/C/D

    const int row = tile_m + idx;
    const float s = scale[row];
    const float* xp = X + (size_t)row * K + 2 * hf;            // contiguous float2 per step
    const float* wp = W + (size_t)(2 * hf) * H_DIM + tile_n + idx;

    v8f acc = {};
    for (int k = 0; k < K; k += 4) {
        v2f a = *(const v2f*)(xp + k);
        a.x *= s; a.y *= s;
        v2f b;
        b.x = wp[(size_t)k * H_DIM];
        b.y = wp[(size_t)(k + 1) * H_DIM];
        acc = __builtin_amdgcn_wmma_f32_16x16x4_f32(
            /*neg_a=*/false, a, /*neg_b=*/false, b,
            /*c_mod=*/(short)0, acc, /*reuse_a=*/false, /*reuse_b=*/false);
    }

    float* yp = Y + (size_t)tile_m * H_DIM + tile_n + idx;
#pragma unroll
    for (int r = 0; r < 8; ++r)
        yp[(size_t)(r + 8 * hf) * H_DIM] = acc[r];
}

// ------------------------------------------------- edge scatter-add (bandwidth bound)
// thread t -> edge e = t>>6, 4-float chunk c = (t&63)*4 ; 64 consecutive
// threads stream one 1KB source row (coalesced), atomic-add into dst row.
__global__ void scatter_add_kernel(const float* __restrict__ Y,
                                   const int* __restrict__ src,
                                   const int* __restrict__ dst,
                                   float* __restrict__ AGG) {
    unsigned long long t = (unsigned long long)blockIdx.x * blockDim.x + threadIdx.x;
    if (t >= (unsigned long long)N_EDGES * (H_DIM / 4)) return;
    int e = (int)(t >> 6);
    int c = ((int)t & 63) << 2;
    int sn = src[e], dn = dst[e];
    const float4 v = *(const float4*)(Y + (size_t)sn * H_DIM + c);
    float* o = AGG + (size_t)dn * H_DIM + c;
    atomicAdd(o + 0, v.x);
    atomicAdd(o + 1, v.y);
    atomicAdd(o + 2, v.z);
    atomicAdd(o + 3, v.w);
}

// h = leaky_relu(agg * norm_dst + bias, 0.01)
__global__ void postproc_kernel(const float* __restrict__ AGG,
                                const float* __restrict__ norm_dst,
                                const float* __restrict__ bias,
                                float* __restrict__ Hout, size_t total) {
    size_t i = (size_t)blockIdx.x * blockDim.x + threadIdx.x;
    if (i < total) {
        int c = (int)(i & (H_DIM - 1));
        int r = (int)(i >> 8);
        float v = AGG[i] * norm_dst[r] + bias[c];
        Hout[i] = v > 0.0f ? v : 0.01f * v;
    }
}

// ------------------------------------------------- column pool (sum + max), two pass
#define POOL_BLOCKS 250
#define POOL_ROWS   400   // 250*400 == 100000
__global__ __launch_bounds__(256)
void pool_partial_kernel(const float* __restrict__ H2,
                         float* __restrict__ psum, float* __restrict__ pmax) {
    int c = threadIdx.x;              // column (coalesced across threads)
    int r0 = blockIdx.x * POOL_ROWS;
    float s = 0.0f, m = -INFINITY;
    for (int r = r0; r < r0 + POOL_ROWS; ++r) {
        float v = H2[(size_t)r * H_DIM + c];
        s += v;
        m = fmaxf(m, v);
    }
    psum[blockIdx.x * H_DIM + c] = s;
    pmax[blockIdx.x * H_DIM + c] = m;
}

__global__ void pool_final_kernel(const float* __restrict__ psum,
                                  const float* __restrict__ pmax,
                                  const float* __restrict__ H2,
                                  const int* __restrict__ node_index,
                                  float* __restrict__ pooled) {
    int c = threadIdx.x;              // 256 threads
    float s = 0.0f, m = -INFINITY;
    for (int b = 0; b < POOL_BLOCKS; ++b) {
        s += psum[b * H_DIM + c];
        m = fmaxf(m, pmax[b * H_DIM + c]);
    }
    pooled[c] = s;
    pooled[H_DIM + c] = m;
    pooled[2 * H_DIM + c] = H2[(size_t)(*node_index) * H_DIM + c];
}

// ------------------------------------------------- head: GEMV + reparam + log_prob
// fd = pooled(768) @ Wg(768x1024) + bg ; mu=fd[:512]; sigma=|fd[512:]|
// feat_new = mu + sigma*eps ; log_prob = mean(-0.5*eps^2 - log(sigma) - 0.5*log2pi)
__global__ __launch_bounds__(512)
void head_kernel(const float* __restrict__ pooled, const float* __restrict__ Wg,
                 const float* __restrict__ bg, const float* __restrict__ eps,
                 float* __restrict__ out) {
    __shared__ float sp[3 * H_DIM];
    __shared__ float red[512];
    int j = threadIdx.x;                      // 0..511
    for (int i = j; i < 3 * H_DIM; i += 512) sp[i] = pooled[i];
    __syncthreads();

    float fmu = bg[j], fsg = bg[512 + j];
    for (int i = 0; i < 3 * H_DIM; ++i) {
        float p = sp[i];
        const float* wrow = Wg + (size_t)i * (2 * F_DIM);
        fmu += p * wrow[j];
        fsg += p * wrow[512 + j];
    }
    float sg = fabsf(fsg);
    float e = eps[j];
    out[j]        = fmu + sg * e;   // feat_new
    out[512 + j]  = fmu;            // mu
    out[1024 + j] = sg;             // sigma

    const float LOG2PI = 1.8378770664093454f;
    red[j] = -0.5f * e * e - logf(sg) - 0.5f * LOG2PI;
    __syncthreads();
    for (int st = 256; st > 0; st >>= 1) {
        if (j < st) red[j] += red[j + st];
        __syncthreads();
    }
    if (j == 0) out[1536] = red[0] / 512.0f;  // log_prob
}

// ---------------------------------------------------------------- launcher
extern "C" void kernel_launch(void* const* d_in, const int* in_sizes, int n_in,
                              void* d_out, int out_size, void* d_ws, size_t ws_size,
                              hipStream_t stream) {
    const float* feat = (const float*)d_in[0];
    const int*   src  = (const int*)d_in[1];
    const int*   dst  = (const int*)d_in[2];
    const int*   nidx = (const int*)d_in[3];
    const float* W1   = (const float*)d_in[4];
    const float* b1   = (const float*)d_in[5];
    const float* W2   = (const float*)d_in[6];
    const float* b2   = (const float*)d_in[7];
    const float* Wg   = (const float*)d_in[8];
    const float* bg   = (const float*)d_in[9];
    const float* eps  = (const float*)d_in[10];

    const size_t NH = (size_t)N_NODES * H_DIM;
    float* ws       = (float*)d_ws;
    float* norm_src = ws;                       // N  (degrees, then rsqrt in place)
    float* norm_dst = norm_src + N_NODES;       // N
    float* bufA     = norm_dst + N_NODES;       // N*H
    float* bufB     = bufA + NH;                // N*H
    float* psum     = bufB + NH;                // 250*256
    float* pmax     = psum + POOL_BLOCKS * H_DIM;
    float* pooled   = pmax + POOL_BLOCKS * H_DIM;   // 768

    // degrees -> symmetric-norm coefficients
    zero_kernel<<<(2 * N_NODES + 255) / 256, 256, 0, stream>>>(norm_src, 2 * N_NODES);
    deg_kernel<<<(N_EDGES + 255) / 256, 256, 0, stream>>>(src, dst, norm_src, norm_dst);
    norm_kernel<<<(N_NODES + 255) / 256, 256, 0, stream>>>(norm_src, norm_dst);

    const int gemm_blocks = (N_NODES / 16) * (H_DIM / 16) / 4;   // 25000
    const int scat_blocks = N_EDGES * (H_DIM / 4) / 256;         // 400000
    const int nh_blocks   = (int)((NH + 255) / 256);

    // layer 1: Y1 = (feat*norm_src) @ W1 ; agg ; leaky(agg*norm_dst + b1)
    gemm_scaled_wmma<<<gemm_blocks, 128, 0, stream>>>(feat, W1, norm_src, bufA, F_DIM);
    zero_kernel<<<nh_blocks, 256, 0, stream>>>(bufB, NH);
    scatter_add_kernel<<<scat_blocks, 256, 0, stream>>>(bufA, src, dst, bufB);
    postproc_kernel<<<nh_blocks, 256, 0, stream>>>(bufB, norm_dst, b1, bufA, NH);

    // layer 2
    gemm_scaled_wmma<<<gemm_blocks, 128, 0, stream>>>(bufA, W2, norm_src, bufB, H_DIM);
    zero_kernel<<<nh_blocks, 256, 0, stream>>>(bufA, NH);
    scatter_add_kernel<<<scat_blocks, 256, 0, stream>>>(bufB, src, dst, bufA);
    postproc_kernel<<<nh_blocks, 256, 0, stream>>>(bufA, norm_dst, b2, bufB, NH);

    // pooled = [sum(h2,0), max(h2,0), h2[node_index]]
    pool_partial_kernel<<<POOL_BLOCKS, 256, 0, stream>>>(bufB, psum, pmax);
    pool_final_kernel<<<1, 256, 0, stream>>>(psum, pmax, bufB, nidx, pooled);

    // head -> d_out = [feat_new(512), mu(512), sigma(512), log_prob(1)]
    head_kernel<<<1, 512, 0, stream>>>(pooled, Wg, bg, eps, (float*)d_out);
}